// DocREDModel_52836687675513
// MI455X (gfx1250) — compile-verified
//
#include <hip/hip_runtime.h>

typedef float v2f __attribute__((ext_vector_type(2)));
typedef float v4f __attribute__((ext_vector_type(4)));
typedef float v8f __attribute__((ext_vector_type(8)));

#define B_  8
#define L_  2048
#define H_  1024
#define E_  32
#define M_  4
#define P_  1024
#define NL  97
#define NLP 112          // 97 padded to 7 WMMA N-frags of 16
#define KP  1028         // LDS row stride (floats): +4 pad -> conflict-free b64 A-frag reads

// workspace layout (float offsets)
#define WS_ENT   0                      // [256][1024]
#define WS_ENTWT 262144                 // [256][1024]  (dense_b folded in)
#define WS_ENTWB 524288                 // [256][1024]
#define WS_OWP   786432                 // [1024][112]  out_w zero-padded
#define WS_OBP   901120                 // [112]        out_b zero-padded

// ---------------------------------------------------------------------------
// Kernel 0: zero-pad out_w [1024,97] -> [1024,112] and out_b -> [112]
// ---------------------------------------------------------------------------
__global__ void pad_outw_kernel(const float* __restrict__ ow,
                                const float* __restrict__ ob,
                                float* __restrict__ owp,
                                float* __restrict__ obp) {
    int idx = blockIdx.x * 256 + threadIdx.x;
    if (idx < H_ * NLP) {
        int k = idx / NLP, n = idx % NLP;
        owp[idx] = (n < NL) ? ow[k * NL + n] : 0.0f;
    }
    if (idx < NLP) obp[idx] = (idx < NL) ? ob[idx] : 0.0f;
}

// ---------------------------------------------------------------------------
// Kernel 1: ent[b,e,:] = sum_m hidden[b, pos[b,e,m], :]
// 256 blocks (one per b*E+e) x 256 threads, float4 per thread. HBM-bound.
// ---------------------------------------------------------------------------
__global__ void __launch_bounds__(256)
ent_sum_kernel(const float* __restrict__ hs, const int* __restrict__ epi,
               float* __restrict__ ent) {
    int be = blockIdx.x;          // b*32 + e
    int b  = be >> 5;
    int t  = threadIdx.x;         // float4 column index
    const int* p = epi + be * M_;
    v4f acc = {0.f, 0.f, 0.f, 0.f};
    #pragma unroll
    for (int m = 0; m < M_; ++m) {
        int pos = p[m];
        const v4f* row = (const v4f*)(hs + (size_t)(b * L_ + pos) * H_);
        acc += row[t];
    }
    ((v4f*)(ent + (size_t)be * H_))[t] = acc;
}

// ---------------------------------------------------------------------------
// Kernel 2: entWt = ent @ W_top + dense_b ; entWb = ent @ W_bot
// f32 WMMA 16x16x4. Block = 256 threads = 8 waves; tile 16(M) x 128(N);
// grid.x = 16 M-tiles * 8 N-tiles = 128, grid.y = half (0=top,1=bot).
// A tile (16x1024) staged in LDS, B frags streamed from dense_w (L2-hot).
// ---------------------------------------------------------------------------
__global__ void __launch_bounds__(256)
gemm_ent_w_kernel(const float* __restrict__ ent, const float* __restrict__ dw,
                  const float* __restrict__ db,
                  float* __restrict__ entWt, float* __restrict__ entWb) {
    __shared__ float Als[16 * KP];
    int mt   = blockIdx.x & 15;   // M tile (16 rows of ent)
    int nt   = blockIdx.x >> 4;   // N tile (128 cols)
    int half = blockIdx.y;        // 0: W_top (+bias), 1: W_bot
    int t    = threadIdx.x;

    // cooperative load of A tile [16 x 1024] into LDS (row-padded)
    #pragma unroll
    for (int i = 0; i < 16; ++i) {
        v4f v = ((const v4f*)(ent + (size_t)(mt * 16 + i) * H_))[t];
        *(v4f*)&Als[i * KP + 4 * t] = v;
    }
    __syncthreads();

    int w  = t >> 5, l = t & 31;
    int lm = l & 15, lh = l >> 4;           // lane -> (M/N index, K-half)
    int n  = nt * 128 + w * 16 + lm;        // output column
    const float* wp = dw + (size_t)half * H_ * H_;   // W half, rows [0,1024)

    v8f c;
    float bias = (half == 0) ? db[n] : 0.0f;
    #pragma unroll
    for (int v = 0; v < 8; ++v) c[v] = bias;

    #pragma unroll 4
    for (int k = 0; k < H_; k += 4) {
        v2f a = *(const v2f*)&Als[lm * KP + k + 2 * lh];     // A[lm][k+2lh .. +1]
        v2f bf;
        bf[0] = wp[(size_t)(k + 2 * lh + 0) * H_ + n];       // B[k'][n]
        bf[1] = wp[(size_t)(k + 2 * lh + 1) * H_ + n];
        c = __builtin_amdgcn_wmma_f32_16x16x4_f32(false, a, false, bf,
                                                  (short)0, c, false, false);
    }

    float* outp = half ? entWb : entWt;
    int mbase = mt * 16;
    #pragma unroll
    for (int v = 0; v < 8; ++v)
        outp[(size_t)(mbase + v + 8 * lh) * H_ + n] = c[v];
}

// ---------------------------------------------------------------------------
// Kernel 3: per 16 pair-rows: x = entWt[s] + entWb[o]  (gather-add, L2-hot),
// then logits = x @ out_w_pad + out_b_pad via f32 WMMA (7 waves, one N-frag
// each covering padded N=112; predicated store of n<97).
// ---------------------------------------------------------------------------
__global__ void __launch_bounds__(224)
rel_logits_kernel(const float* __restrict__ entWt, const float* __restrict__ entWb,
                  const float* __restrict__ owp, const float* __restrict__ obp,
                  const int* __restrict__ ht, float* __restrict__ out) {
    __shared__ float Xls[16 * KP];
    int r0 = blockIdx.x * 16;     // 16 pair rows; all share batch b
    int b  = r0 >> 10;
    int t  = threadIdx.x;

    // build x tile [16 x 1024] = entWt[s] + entWb[o] into LDS (4096 float4s)
    for (int idx = t; idx < 4096; idx += 224) {
        int row = idx >> 8, c4 = idx & 255;
        int r = r0 + row;
        int s = ht[2 * r], o = ht[2 * r + 1];
        v4f xa = ((const v4f*)(entWt + (size_t)(b * E_ + s) * H_))[c4];
        v4f xb = ((const v4f*)(entWb + (size_t)(b * E_ + o) * H_))[c4];
        v4f x  = xa + xb;
        *(v4f*)&Xls[row * KP + 4 * c4] = x;
    }
    __syncthreads();

    int w  = t >> 5, l = t & 31;
    int lm = l & 15, lh = l >> 4;
    int n  = w * 16 + lm;                   // padded column in [0,112)

    v8f c;
    float bias = obp[n];
    #pragma unroll
    for (int v = 0; v < 8; ++v) c[v] = bias;

    #pragma unroll 4
    for (int k = 0; k < H_; k += 4) {
        v2f a = *(const v2f*)&Xls[lm * KP + k + 2 * lh];
        v2f bf;
        bf[0] = owp[(size_t)(k + 2 * lh + 0) * NLP + n];
        bf[1] = owp[(size_t)(k + 2 * lh + 1) * NLP + n];
        c = __builtin_amdgcn_wmma_f32_16x16x4_f32(false, a, false, bf,
                                                  (short)0, c, false, false);
    }

    if (n < NL) {
        #pragma unroll
        for (int v = 0; v < 8; ++v)
            out[(size_t)(r0 + v + 8 * lh) * NL + n] = c[v];
    }
}

// ---------------------------------------------------------------------------
extern "C" void kernel_launch(void* const* d_in, const int* in_sizes, int n_in,
                              void* d_out, int out_size, void* d_ws, size_t ws_size,
                              hipStream_t stream) {
    const float* hs  = (const float*)d_in[0];   // hidden_states [8,2048,1024]
    const float* dw  = (const float*)d_in[1];   // dense_w [2048,1024]
    const float* db  = (const float*)d_in[2];   // dense_b [1024]
    const float* ow  = (const float*)d_in[3];   // out_w [1024,97]
    const float* ob  = (const float*)d_in[4];   // out_b [97]
    const int*   epi = (const int*)d_in[5];     // entity_position_ids [8,32,4]
    const int*   ht  = (const int*)d_in[6];     // head_tail_idxs [8,1024,2]
    float* ws  = (float*)d_ws;
    float* out = (float*)d_out;

    float* ent   = ws + WS_ENT;
    float* entWt = ws + WS_ENTWT;
    float* entWb = ws + WS_ENTWB;
    float* owp   = ws + WS_OWP;
    float* obp   = ws + WS_OBP;

    pad_outw_kernel<<<(H_ * NLP + 255) / 256, 256, 0, stream>>>(ow, ob, owp, obp);
    ent_sum_kernel<<<B_ * E_, 256, 0, stream>>>(hs, epi, ent);
    gemm_ent_w_kernel<<<dim3(128, 2), 256, 0, stream>>>(ent, dw, db, entWt, entWb);
    rel_logits_kernel<<<(B_ * P_) / 16, 224, 0, stream>>>(entWt, entWb, owp, obp, ht, out);
}